// softAttention_59708635349628
// MI455X (gfx1250) — compile-verified
//
#include <hip/hip_runtime.h>

#define B_ 32
#define S_ 2048
#define H_ 1024
#define NB 8                                   // blocks per batch, pass 1
#define WPB 8                                  // waves per block (256 thr)
#define ROWS_PER_WAVE (S_ / (NB * WPB))        // 32
#define PARTIALS NB                            // per-batch partials in ws
#define PB_STRIDE (PARTIALS * H_ + 2 * PARTIALS) // floats per batch in ws

typedef __attribute__((ext_vector_type(2))) float v2f;
typedef __attribute__((ext_vector_type(8))) float v8f;

// ---------------------------------------------------------------------------
// Pass 1: one-pass online-softmax attention. Each wave owns 32 rows of one
// batch; row (1024 f32) = 8 x float4 per lane, identical layout for the
// score dot-product and the context accumulation -> encoder read EXACTLY once.
// ---------------------------------------------------------------------------
__global__ __launch_bounds__(256) void attn_pass1(
    const float* __restrict__ enc, const float* __restrict__ dec,
    const float* __restrict__ mask, float* __restrict__ ws)
{
    __shared__ float red_m[WPB];
    __shared__ float red_s[WPB];
    __shared__ float ctxbuf[WPB * H_];

    const int b      = blockIdx.x / NB;
    const int blkInB = blockIdx.x % NB;
    const int tid    = threadIdx.x;
    const int wv     = tid >> 5;
    const int lane   = tid & 31;
    const int s0     = (blkInB * WPB + wv) * ROWS_PER_WAVE;

    // decoder fragment (kept in regs for all 32 rows): h = k*128 + lane*4
    const float4* dec4 = (const float4*)(dec + (size_t)b * H_);
    float4 dfrag[8];
#pragma unroll
    for (int k = 0; k < 8; ++k) dfrag[k] = dec4[k * 32 + lane];

    // one mask value per lane, broadcast per-row via shfl
    const float maskv = mask[(size_t)b * S_ + s0 + lane];

    float4 ctx[8];
#pragma unroll
    for (int k = 0; k < 8; ++k) ctx[k] = make_float4(0.f, 0.f, 0.f, 0.f);
    float m = -1e30f, ssum = 0.f;

    const float4* encB = (const float4*)(enc + (size_t)b * S_ * H_);
    for (int r = 0; r < ROWS_PER_WAVE; ++r) {
        const float4* row4 = encB + (size_t)(s0 + r) * (H_ / 4);
        float4 row[8];
#pragma unroll
        for (int k = 0; k < 8; ++k) row[k] = row4[k * 32 + lane];

        float d = 0.f;
#pragma unroll
        for (int k = 0; k < 8; ++k)
            d += row[k].x * dfrag[k].x + row[k].y * dfrag[k].y +
                 row[k].z * dfrag[k].z + row[k].w * dfrag[k].w;
#pragma unroll
        for (int off = 16; off >= 1; off >>= 1) d += __shfl_xor(d, off, 32);

        const float logit = d * __shfl(maskv, r, 32);
        const float mn = fmaxf(m, logit);
        const float c  = __expf(m - mn);       // rescale old accumulators
        const float p  = __expf(logit - mn);
        ssum = ssum * c + p;
#pragma unroll
        for (int k = 0; k < 8; ++k) {
            ctx[k].x = ctx[k].x * c + p * row[k].x;
            ctx[k].y = ctx[k].y * c + p * row[k].y;
            ctx[k].z = ctx[k].z * c + p * row[k].z;
            ctx[k].w = ctx[k].w * c + p * row[k].w;
        }
        m = mn;
    }

    // ---- combine the 8 wave partials of this block through LDS ----
    if (lane == 0) red_m[wv] = m;
    __syncthreads();
    float mB = red_m[0];
#pragma unroll
    for (int i = 1; i < WPB; ++i) mB = fmaxf(mB, red_m[i]);
    const float w = __expf(m - mB);
    if (lane == 0) red_s[wv] = ssum * w;
    float4* cb = (float4*)(ctxbuf + wv * H_);
#pragma unroll
    for (int k = 0; k < 8; ++k) {
        float4 v = ctx[k];
        v.x *= w; v.y *= w; v.z *= w; v.w *= w;
        cb[k * 32 + lane] = v;
    }
    __syncthreads();

    float sB = 0.f;
#pragma unroll
    for (int i = 0; i < WPB; ++i) sB += red_s[i];
    float4 acc = make_float4(0.f, 0.f, 0.f, 0.f);
#pragma unroll
    for (int i = 0; i < WPB; ++i) {
        float4 v = ((const float4*)(ctxbuf + i * H_))[tid];
        acc.x += v.x; acc.y += v.y; acc.z += v.z; acc.w += v.w;
    }
    float* base = ws + (size_t)b * PB_STRIDE;
    ((float4*)(base + blkInB * H_))[tid] = acc;
    if (tid == 0) {
        base[PARTIALS * H_ + blkInB] = mB;
        base[PARTIALS * H_ + PARTIALS + blkInB] = sB;
    }
}

// ---------------------------------------------------------------------------
// Pass 2: merge the 8 per-block partials per batch, normalize, emit context.
// ---------------------------------------------------------------------------
__global__ __launch_bounds__(256) void attn_combine(
    const float* __restrict__ ws, float* __restrict__ ctx_out)
{
    const int b = blockIdx.x, tid = threadIdx.x;
    const float* base = ws + (size_t)b * PB_STRIDE;
    const float* ms = base + PARTIALS * H_;
    const float* ss = ms + PARTIALS;

    float M = ms[0];
#pragma unroll
    for (int i = 1; i < PARTIALS; ++i) M = fmaxf(M, ms[i]);
    float e[PARTIALS], T = 0.f;
#pragma unroll
    for (int i = 0; i < PARTIALS; ++i) { e[i] = __expf(ms[i] - M); T += e[i] * ss[i]; }
    const float inv = 1.f / T;

    float4 acc = make_float4(0.f, 0.f, 0.f, 0.f);
#pragma unroll
    for (int i = 0; i < PARTIALS; ++i) {
        float4 v = ((const float4*)(base + i * H_))[tid];
        acc.x += e[i] * v.x; acc.y += e[i] * v.y;
        acc.z += e[i] * v.z; acc.w += e[i] * v.w;
    }
    acc.x *= inv; acc.y *= inv; acc.z *= inv; acc.w *= inv;
    ((float4*)(ctx_out + (size_t)b * H_))[tid] = acc;
}

// ---------------------------------------------------------------------------
// Pass 3: out = tanh(x @ W^T + bias) with V_WMMA_F32_16X16X4_F32.
// x replicates torch.stack((ctx,dec)).view(B,-1):
//   b<16 : x[b] = [ctx[2b] | ctx[2b+1]];  b>=16 : x[b] = [dec[2b-32] | dec[2b-31]]
// Each wave computes one 16x16 tile; K = 2048 in steps of 4.
// ---------------------------------------------------------------------------
__global__ __launch_bounds__(128) void attn_out_wmma(
    const float* __restrict__ ctx, const float* __restrict__ dec,
    const float* __restrict__ W, const float* __restrict__ bias,
    float* __restrict__ out)
{
    const int wave = blockIdx.x * (blockDim.x >> 5) + (threadIdx.x >> 5);
    const int lane = threadIdx.x & 31;
    const int m0 = (wave >> 6) << 4;   // 0 or 16 (row tile)
    const int n0 = (wave & 63) << 4;   // output-column tile
    const int lm = lane & 15;
    const int hi = lane >> 4;          // A/B hold K = k0 + 2*hi + {0,1}

    const float* src = (m0 == 0) ? ctx : dec;          // homogeneous tile
    const float* xlo = src + (size_t)(2 * lm) * H_;    // x[m][j],   j <  1024
    const float* xhi = xlo + H_;                       // x[m][j],   j >= 1024
    const float* wrow = W + (size_t)(n0 + lm) * (2 * H_);

    v8f acc = {};
#pragma unroll 4
    for (int k0 = 0; k0 < 1024; k0 += 4) {
        const int kk = k0 + 2 * hi;
        v2f a  = { xlo[kk],  xlo[kk + 1]  };
        v2f bm = { wrow[kk], wrow[kk + 1] };
        acc = __builtin_amdgcn_wmma_f32_16x16x4_f32(
            false, a, false, bm, (short)0, acc, false, false);
    }
#pragma unroll 4
    for (int k0 = 1024; k0 < 2048; k0 += 4) {
        const int kk = k0 + 2 * hi;
        v2f a  = { xhi[kk - 1024], xhi[kk - 1023] };
        v2f bm = { wrow[kk],       wrow[kk + 1]   };
        acc = __builtin_amdgcn_wmma_f32_16x16x4_f32(
            false, a, false, bm, (short)0, acc, false, false);
    }

    const int col = n0 + lm;
    const float bv = bias[col];
#pragma unroll
    for (int r = 0; r < 8; ++r) {
        const int row = m0 + r + (hi << 3);   // C/D layout: lanes>=16 -> M+8
        out[(size_t)row * H_ + col] = tanhf(acc[r] + bv);
    }
}

// ---------------------------------------------------------------------------
extern "C" void kernel_launch(void* const* d_in, const int* in_sizes, int n_in,
                              void* d_out, int out_size, void* d_ws, size_t ws_size,
                              hipStream_t stream) {
    const float* enc  = (const float*)d_in[0];  // [B,S,H]
    const float* dec  = (const float*)d_in[1];  // [B,H]
    const float* mask = (const float*)d_in[2];  // [B,S]
    const float* W    = (const float*)d_in[3];  // [H,2H]
    const float* bias = (const float*)d_in[4];  // [H]

    float* out = (float*)d_out;        // [0, B*H)      : tanh output
    float* ctx = out + B_ * H_;        // [B*H, 2*B*H)  : context_vector
    float* ws  = (float*)d_ws;         // ~1.0 MB of partials

    attn_pass1 <<<dim3(B_ * NB), dim3(256), 0, stream>>>(enc, dec, mask, ws);
    attn_combine<<<dim3(B_),      dim3(256), 0, stream>>>(ws, ctx);
    attn_out_wmma<<<dim3(32),     dim3(128), 0, stream>>>(ctx, dec, W, bias, out);
}